// RouteNet_7121055776897
// MI455X (gfx1250) — compile-verified
//
#include <hip/hip_runtime.h>
#include <hip/hip_bf16.h>

// ---------------------------------------------------------------------------
// RouteNet fused forward for MI455X (gfx1250, wave32).
//
// One wave owns a 16-row batch tile end-to-end: input GEMM -> 7 DAG layer
// sweeps -> output log_softmax, all activations LDS-resident. All matrix math
// uses V_WMMA_F32_16X16X4_F32 (fp32 exactness: gate>0 thresholds, softmax).
// Phase 1 stages each x chunk ONCE and accumulates all 8 input banks in
// registers (8 x v8f accumulators), eliminating 8x redundant x traffic.
// ---------------------------------------------------------------------------

typedef __attribute__((ext_vector_type(2))) float v2f;
typedef __attribute__((ext_vector_type(8))) float v8f;

#define N_LAYERS 8
#define BPL      8
#define DD       10      // neurons per bank
#define NIN      784
#define NOUTC    10
#define BSZ      32768
#define EPS      (BPL * BPL)             // edges per layer step (64)
#define NEDGE    ((N_LAYERS - 1) * EPS)  // 448
#define TK       112                     // K-chunk for input GEMM (784 = 7*112)

__device__ __forceinline__ v8f wmma4_f32(v2f a, v2f b, v8f c) {
    // 8 args: (neg_a, A, neg_b, B, c_mod, C, reuse_a, reuse_b)
    return __builtin_amdgcn_wmma_f32_16x16x4_f32(
        false, a, false, b, (short)0, c, false, false);
}

__global__ __launch_bounds__(32) void routenet_fused(
    const float* __restrict__ x,       // [B][784]
    const float* __restrict__ in_W,    // [8][784][10]
    const float* __restrict__ in_b,    // [8][10]
    const float* __restrict__ gate_W,  // [448][10]
    const float* __restrict__ gate_b,  // [448]
    const float* __restrict__ data_W,  // [448][10][10]
    const float* __restrict__ out_W,   // [8][10][10]
    float* __restrict__ out_logits,    // [B][10]
    float* __restrict__ out_totgate,   // [B]
    float* __restrict__ out_gstatus,   // [B][64][64] (pre-zeroed)
    unsigned* __restrict__ open_cnt)   // global open-gate counter (pre-zeroed)
{
    __shared__ float sX[16][TK + 1];        // x chunk (padded vs bank conflicts)
    __shared__ float sW[TK][16];            // weight chunk, B-frag friendly
    __shared__ float sA[2][BPL][16][17];    // ping-pong activations, zero-padded
    __shared__ float sGate[16];
    __shared__ float sOut[16][17];
    __shared__ unsigned sCnt[16];

    const int lane = threadIdx.x;       // 0..31 (wave32)
    const int ln   = lane & 15;         // A-frag row M / B,C-frag col N
    const int half = lane >> 4;         // 0 or 1
    const int klo  = half * 2;          // K offset for A/B fragments
    const int b0   = blockIdx.x * 16;   // batch tile base

    // ---------------- Phase 1: input banks  acts0 = relu(x @ in_W + b) ------
    // kc outer / bank inner: stage each x chunk once, hold 8 accumulators.
    v8f acc8[BPL] = {};                 // 64 VGPRs of accumulators

    for (int kc = 0; kc < NIN; kc += TK) {
        // stage x tile chunk [16][TK] exactly once per kc
        for (int idx = lane; idx < 16 * TK; idx += 32) {
            int r = idx / TK, c = idx - r * TK;
            sX[r][c] = x[(size_t)(b0 + r) * NIN + kc + c];
        }
        __syncthreads();
        if (kc + TK < NIN)  // hint prefetch of next x chunk (global_prefetch_b8)
            __builtin_prefetch(&x[(size_t)(b0 + ln) * NIN + kc + TK], 0, 1);
#pragma unroll
        for (int bank = 0; bank < BPL; ++bank) {
            // stage this bank's W chunk [TK][16], cols >= 10 zero-padded
            for (int idx = lane; idx < TK * 16; idx += 32) {
                int kk = idx >> 4, n = idx & 15;
                sW[kk][n] = (n < DD)
                    ? in_W[((size_t)bank * NIN + (kc + kk)) * DD + n] : 0.0f;
            }
            __syncthreads();
#pragma unroll
            for (int kk = 0; kk < TK; kk += 4) {
                v2f a, b;
                a.x = sX[ln][kk + klo];     a.y = sX[ln][kk + klo + 1];
                b.x = sW[kk + klo][ln];     b.y = sW[kk + klo + 1][ln];
                acc8[bank] = wmma4_f32(a, b, acc8[bank]);
            }
            __syncthreads();   // before next bank (or next kc) overwrites LDS
        }
    }
#pragma unroll
    for (int bank = 0; bank < BPL; ++bank) {
        float bias = (ln < DD) ? in_b[bank * DD + ln] : 0.0f;
#pragma unroll
        for (int j = 0; j < 8; ++j) {
            float v = acc8[bank][j] + bias;     // element [M=j+8*half][N=ln]
            v = v > 0.0f ? v : 0.0f;
            if (ln >= DD) v = 0.0f;             // keep padding exactly zero
            sA[0][bank][half * 8 + j][ln] = v;
        }
    }
    __syncthreads();

    // ---------------- Phase 2: 7 layer sweeps over the 8x8 edge groups ------
    float    tg    = 0.0f;   // per-row total gate (lanes 0..15)
    unsigned opens = 0u;     // per-row open count (lanes 0..15)
    int cur = 0;

    for (int l = 0; l < N_LAYERS - 1; ++l) {
        for (int t = 0; t < BPL; ++t) {
            v8f acc = {};
            for (int s = 0; s < BPL; ++s) {
                const int e = l * EPS + s * BPL + t;
                if (lane < 16) {
                    float dot = gate_b[e];
#pragma unroll
                    for (int d = 0; d < DD; ++d)
                        dot += sA[cur][s][ln][d] * gate_W[e * DD + d];
                    float g = dot < 0.0f ? 0.0f : (dot > 1.0f ? 1.0f : dot);
                    sGate[ln] = g;
                    tg += g;
                    opens += (g > 0.0f) ? 1u : 0u;
                    const int srcg = l * BPL + s, tgtg = (l + 1) * BPL + t;
                    out_gstatus[((size_t)(b0 + ln) * 64 + srcg) * 64 + tgtg] =
                        (g > 0.0f) ? 1.0f : 0.0f;
                }
                __syncthreads();
                const float gr = sGate[ln];   // gate for A-frag row M=ln
                // gated matvec: (g[b]*s[b,:]) @ data_W[e]  (K = 10, 3 chunks)
#pragma unroll
                for (int kk = 0; kk < 12; kk += 4) {
                    const int k0 = kk + klo, k1 = k0 + 1;
                    v2f a, b;
                    a.x = sA[cur][s][ln][k0] * gr;   // cols 10..15 are zero
                    a.y = sA[cur][s][ln][k1] * gr;
                    b.x = (k0 < DD && ln < DD)
                        ? data_W[((size_t)e * DD + k0) * DD + ln] : 0.0f;
                    b.y = (k1 < DD && ln < DD)
                        ? data_W[((size_t)e * DD + k1) * DD + ln] : 0.0f;
                    acc = wmma4_f32(a, b, acc);
                }
                __syncthreads();
            }
            // relu, store into next-layer target bank
#pragma unroll
            for (int j = 0; j < 8; ++j) {
                float v = acc[j];
                sA[cur ^ 1][t][half * 8 + j][ln] = v > 0.0f ? v : 0.0f;
            }
        }
        __syncthreads();
        cur ^= 1;
    }

    // ---------------- Phase 3: output heads + log_softmax -------------------
    v8f acc = {};
    for (int bank = 0; bank < BPL; ++bank) {
#pragma unroll
        for (int kk = 0; kk < 12; kk += 4) {
            const int k0 = kk + klo, k1 = k0 + 1;
            v2f a, b;
            a.x = sA[cur][bank][ln][k0];
            a.y = sA[cur][bank][ln][k1];
            b.x = (k0 < DD && ln < NOUTC)
                ? out_W[((size_t)bank * DD + k0) * NOUTC + ln] : 0.0f;
            b.y = (k1 < DD && ln < NOUTC)
                ? out_W[((size_t)bank * DD + k1) * NOUTC + ln] : 0.0f;
            acc = wmma4_f32(a, b, acc);
        }
    }
#pragma unroll
    for (int j = 0; j < 8; ++j) sOut[half * 8 + j][ln] = acc[j];
    if (lane < 16) sCnt[ln] = opens;
    __syncthreads();

    if (lane < 16) {
        float m = sOut[ln][0];
        for (int o = 1; o < NOUTC; ++o) m = fmaxf(m, sOut[ln][o]);
        float ssum = 0.0f;
        for (int o = 0; o < NOUTC; ++o) ssum += expf(sOut[ln][o] - m);
        const float lse = m + logf(ssum);
        for (int o = 0; o < NOUTC; ++o)
            out_logits[(size_t)(b0 + ln) * NOUTC + o] = sOut[ln][o] - lse;
        out_totgate[b0 + ln] = tg;
    }
    if (lane == 0) {
        unsigned tot = 0;
        for (int i = 0; i < 16; ++i) tot += sCnt[i];
        atomicAdd(open_cnt, tot);   // integer: deterministic
    }
}

__global__ void routenet_finalize(const unsigned* __restrict__ cnt,
                                  float* __restrict__ prob_out) {
    *prob_out = (float)(*cnt) / (float)(NEDGE * BSZ);
}

// ---------------------------------------------------------------------------
extern "C" void kernel_launch(void* const* d_in, const int* in_sizes, int n_in,
                              void* d_out, int out_size, void* d_ws, size_t ws_size,
                              hipStream_t stream) {
    const float* x      = (const float*)d_in[0];
    const float* in_W   = (const float*)d_in[1];
    const float* in_b   = (const float*)d_in[2];
    const float* gate_W = (const float*)d_in[3];
    const float* gate_b = (const float*)d_in[4];
    const float* data_W = (const float*)d_in[5];
    const float* out_W  = (const float*)d_in[6];
    // d_in[7]/d_in[8] (edge_src/edge_tgt) follow the fixed DAG; indices are
    // recomputed analytically in-kernel.

    float* out         = (float*)d_out;
    float* out_logits  = out;                              // [B][10]
    float* out_totgate = out + (size_t)BSZ * NOUTC;        // [B]
    float* out_prob    = out + (size_t)BSZ * (NOUTC + 1);  // scalar
    float* out_gstat   = out_prob + 1;                     // [B][64][64]
    unsigned* cnt      = (unsigned*)d_ws;

    // Per-call zeroing (graph-capturable): open counter + full gate_status
    // region (only 448 of 4096 (src,tgt) pairs are ever written by the kernel).
    hipMemsetAsync(cnt, 0, sizeof(unsigned), stream);
    hipMemsetAsync(out_gstat, 0, (size_t)BSZ * 64 * 64 * sizeof(float), stream);

    routenet_fused<<<BSZ / 16, 32, 0, stream>>>(
        x, in_W, in_b, gate_W, gate_b, data_W, out_W,
        out_logits, out_totgate, out_gstat, cnt);
    routenet_finalize<<<1, 1, 0, stream>>>(cnt, out_prob);
}